// CSIDecoder_38749194945149
// MI455X (gfx1250) — compile-verified
//
#include <hip/hip_runtime.h>
#include <hip/hip_bf16.h>

typedef float v2f __attribute__((ext_vector_type(2)));
typedef float v8f __attribute__((ext_vector_type(8)));

#define DMODEL 512
#define CHUNKS 128          // DMODEL / 4
#define WAVES_PER_BLOCK 8
#define ROWS_PER_WAVE 16

// Fused LayerNorm(512) + Linear(512->2) + subcarrier scaling, single pass over x.
// Per 16-row tile: C(16x16) += A(16x4) * B(4x16) with
//   A = V chunks (row0 = ones, row1 = gamma*W0, row2 = gamma*W1, rest 0) from LDS
//   B = x chunk, produced directly by one global_load_b64 per lane (layout match)
// Accumulator lanes 0..15 of VGPR 0/1/2 then hold S1/Y0/Y1 per row.
__global__ void __launch_bounds__(256) csi_decoder_kernel(
    const float* __restrict__ x,
    const float* __restrict__ gamma,
    const float* __restrict__ beta,
    const float* __restrict__ W,       // [2][512]
    const float* __restrict__ bias,    // [2]
    const float* __restrict__ scalers, // [64]
    float* __restrict__ out)           // [N][2] interleaved complex64
{
    __shared__ float Abuf[CHUNKS * 64];  // [chunk][lane][2] floats = 32 KB
    __shared__ float Gsh[4];             // G1_0, G1_1, G0_0+b0, G0_1+b1

    const int tid  = threadIdx.x;
    const int lane = tid & 31;
    const int wave = tid >> 5;

    // ---- Build WMMA A-operand table in LDS (once per block) ----
    // A layout (f32 16x16x4): VGPR v, lane-group g=lane>>4, m=lane&15:
    //   element (M=m, K=4c+2g+v)
    for (int e = tid; e < CHUNKS * 32; e += 256) {
        int c = e >> 5;
        int l = e & 31;
        int m = l & 15;
        int g = l >> 4;
        int k = 4 * c + 2 * g;
        float a0 = 0.0f, a1 = 0.0f;
        if (m == 0)      { a0 = 1.0f;                    a1 = 1.0f; }
        else if (m == 1) { a0 = gamma[k] * W[k];          a1 = gamma[k + 1] * W[k + 1]; }
        else if (m == 2) { a0 = gamma[k] * W[DMODEL + k]; a1 = gamma[k + 1] * W[DMODEL + k + 1]; }
        Abuf[e * 2 + 0] = a0;
        Abuf[e * 2 + 1] = a1;
    }

    // ---- Scalar constants G1_o = sum(gamma*W_o), G0_o = sum(beta*W_o) (wave 0) ----
    if (wave == 0) {
        float g10 = 0.f, g11 = 0.f, g00 = 0.f, g01 = 0.f;
        for (int k = lane; k < DMODEL; k += 32) {
            float ga = gamma[k], be = beta[k];
            float w0 = W[k], w1 = W[DMODEL + k];
            g10 = fmaf(ga, w0, g10);
            g11 = fmaf(ga, w1, g11);
            g00 = fmaf(be, w0, g00);
            g01 = fmaf(be, w1, g01);
        }
        #pragma unroll
        for (int off = 16; off > 0; off >>= 1) {
            g10 += __shfl_xor(g10, off, 32);
            g11 += __shfl_xor(g11, off, 32);
            g00 += __shfl_xor(g00, off, 32);
            g01 += __shfl_xor(g01, off, 32);
        }
        if (lane == 0) {
            Gsh[0] = g10;
            Gsh[1] = g11;
            Gsh[2] = g00 + bias[0];
            Gsh[3] = g01 + bias[1];
        }
    }
    __syncthreads();

    // ---- Main streaming loop: one 16-row x tile per wave ----
    const long tile = (long)blockIdx.x * WAVES_PER_BLOCK + wave;
    const long row0 = tile * ROWS_PER_WAVE;
    const int  n = lane & 15;      // row within tile (B-operand N index)
    const int  g = lane >> 4;      // K-pair selector within chunk
    const float* __restrict__ xr = x + (row0 + n) * (long)DMODEL + 2 * g;

    v8f  acc = {0.f, 0.f, 0.f, 0.f, 0.f, 0.f, 0.f, 0.f};
    float ssq = 0.0f;

    #pragma unroll 4
    for (int c = 0; c < CHUNKS; ++c) {
        // B operand: x[row n][4c + 2g + {0,1}] -> exactly the 4x16 B layout
        v2f xv = __builtin_nontemporal_load((const v2f*)(xr + 4 * c));
        // A operand from LDS: conflict-free (32 lanes x 2 dwords = 64 banks)
        v2f av = *(const v2f*)(&Abuf[c * 64 + lane * 2]);
        // sumsq (only reduction not expressible with a fixed matmul)
        ssq = fmaf(xv.x, xv.x, ssq);
        ssq = fmaf(xv.y, xv.y, ssq);
        // C += A*B : accumulates S1 (M=0), Y0 (M=1), Y1 (M=2) per row (N)
        acc = __builtin_amdgcn_wmma_f32_16x16x4_f32(
            false, av, false, xv, (short)0, acc, false, false);
    }

    // lane l covers k%4 in {2g, 2g+1} of row (l&15): fold halves
    float S2 = ssq + __shfl_xor(ssq, 16, 32);

    // ---- Epilogue: lanes 0..15 each finish one row ----
    if (lane < 16) {
        const float inv_d = 1.0f / (float)DMODEL;
        float S1 = acc[0];   // C[M=0, N=lane]
        float Y0 = acc[1];   // C[M=1, N=lane]
        float Y1 = acc[2];   // C[M=2, N=lane]

        float mu  = S1 * inv_d;
        float var = fmaf(S2, inv_d, -(mu * mu));
        float inv = rsqrtf(var + 1e-5f);

        long row = row0 + lane;
        float scale = fabsf(scalers[row & 63]) + 0.1f;

        float cf0 = fmaf(inv, fmaf(-mu, Gsh[0], Y0), Gsh[2]);
        float cf1 = fmaf(inv, fmaf(-mu, Gsh[1], Y1), Gsh[3]);

        v2f o;
        o.x = cf0 * scale;
        o.y = cf1 * scale;
        __builtin_nontemporal_store(o, (v2f*)(out + 2 * row));
    }
}

extern "C" void kernel_launch(void* const* d_in, const int* in_sizes, int n_in,
                              void* d_out, int out_size, void* d_ws, size_t ws_size,
                              hipStream_t stream) {
    const float* x       = (const float*)d_in[0];
    const float* gamma   = (const float*)d_in[1];
    const float* beta    = (const float*)d_in[2];
    const float* W       = (const float*)d_in[3];
    const float* bias    = (const float*)d_in[4];
    const float* scalers = (const float*)d_in[5];
    float* out = (float*)d_out;

    long nrows  = (long)in_sizes[0] / DMODEL;                 // 262144
    int  blocks = (int)(nrows / (ROWS_PER_WAVE * WAVES_PER_BLOCK)); // 2048

    csi_decoder_kernel<<<blocks, 256, 0, stream>>>(x, gamma, beta, W, bias, scalers, out);
}